// SparseLinear_63496796504161
// MI455X (gfx1250) — compile-verified
//
#include <hip/hip_runtime.h>

// ---------------------------------------------------------------------------
// SparseLinear on MI455X (gfx1250):
//   1) zero dense W (64MB) in workspace
//   2) scatter COO (values,rows,cols) with fp32 global atomics
//   3) dense GEMM out = x @ W^T + bias using V_WMMA_F32_16X16X4_F32,
//      double-buffered through LDS so global_load_b128 for tile k+1
//      overlaps the WMMA stream for tile k.
// ---------------------------------------------------------------------------

typedef float v2f __attribute__((ext_vector_type(2)));
typedef float v8f __attribute__((ext_vector_type(8)));

#define KDIM 4096   // in_features
#define NDIM 4096   // out_features
#define LDSS 36     // LDS row stride in floats (32 data + 4 skew, keeps 16B align)
#define BUFSZ (64 * LDSS)

// ---------------------------------------------------------------------------
// Kernel 1: zero the dense weight matrix (16M floats) with 128-bit stores.
// ---------------------------------------------------------------------------
__global__ __launch_bounds__(256) void sl_zero(float4* __restrict__ w4, int n4) {
  int i = blockIdx.x * blockDim.x + threadIdx.x;
  if (i < n4) w4[i] = make_float4(0.f, 0.f, 0.f, 0.f);
}

// ---------------------------------------------------------------------------
// Kernel 2: COO scatter with native fp32 atomic add (duplicates accumulate).
// ---------------------------------------------------------------------------
__global__ __launch_bounds__(256) void sl_scatter(const float* __restrict__ vals,
                                                  const int* __restrict__ rows,
                                                  const int* __restrict__ cols,
                                                  float* __restrict__ W, int nnz) {
  int i = blockIdx.x * blockDim.x + threadIdx.x;
  if (i < nnz) {
    unsafeAtomicAdd(&W[(size_t)rows[i] * KDIM + cols[i]], vals[i]);
  }
}

// ---------------------------------------------------------------------------
// Kernel 3: tiled WMMA GEMM. Workgroup = 128 threads = 4 waves (wave32).
// Workgroup computes a 64(M) x 64(N) tile of out; each wave a 32x32 subtile
// as 2x2 V_WMMA_F32_16X16X4_F32 accumulators. K blocked by 32, double-
// buffered through LDS with register prefetch of the next global tile.
//
// A-fragment (16x4 f32, 2 VGPRs): lane<16 -> M=lane,    K={0,1} in v0,v1
//                                 lane>=16 -> M=lane-16, K={2,3} in v0,v1
// B-fragment (4x16 f32, 2 VGPRs): lane<16 -> N=lane,    K={0,1}
//                                 lane>=16 -> N=lane-16, K={2,3}
// C/D (16x16 f32, 8 VGPRs): VGPR r holds M = r + 8*(lane>=16), N = lane%16.
// ---------------------------------------------------------------------------
__global__ __launch_bounds__(128) void sl_gemm(const float* __restrict__ x,
                                               const float* __restrict__ W,
                                               const float* __restrict__ bias,
                                               float* __restrict__ out) {
  __shared__ float xs[2][BUFSZ];
  __shared__ float ws[2][BUFSZ];

  const int tid    = threadIdx.x;
  const int wave   = tid >> 5;        // 0..3
  const int lane   = tid & 31;
  const int lane16 = lane & 15;
  const int half   = lane >> 4;       // 0: K lo / M lo-rows ; 1: K hi / M hi-rows

  const int m0 = blockIdx.y << 6;     // batch-row tile origin
  const int n0 = blockIdx.x << 6;     // out-feature tile origin
  const int wm = (wave >> 1) << 5;    // wave row offset inside 64x64 tile
  const int wn = (wave & 1) << 5;     // wave col offset inside 64x64 tile

  v8f acc[2][2];
#pragma unroll
  for (int i = 0; i < 2; ++i)
#pragma unroll
    for (int j = 0; j < 2; ++j)
#pragma unroll
      for (int r = 0; r < 8; ++r) acc[i][j][r] = 0.f;

  // Per-thread staging coordinates: 512 float4 per 64x32 tile / 128 thr = 4 ea.
  int    ldso[4];     // LDS float offset inside a buffer
  size_t gxo[4];      // global float offset for x (add kb)
  size_t gwo[4];      // global float offset for W (add kb)
#pragma unroll
  for (int p = 0; p < 4; ++p) {
    const int idx = tid + (p << 7);
    const int row = idx >> 3;
    const int c4  = (idx & 7) << 2;
    ldso[p] = row * LDSS + c4;
    gxo[p]  = (size_t)(m0 + row) * KDIM + c4;
    gwo[p]  = (size_t)(n0 + row) * KDIM + c4;
  }

  // Per-lane fragment read offsets inside a buffer.
  const int offA = (wm + lane16) * LDSS + (half << 1);
  const int offB = (wn + lane16) * LDSS + (half << 1);

  float4 rx[4], rw[4];

  // Prologue: fetch tile kb=0 and stage into buffer 0.
#pragma unroll
  for (int p = 0; p < 4; ++p) {
    rx[p] = *(const float4*)&x[gxo[p]];
    rw[p] = *(const float4*)&W[gwo[p]];
  }
#pragma unroll
  for (int p = 0; p < 4; ++p) {
    *(float4*)&xs[0][ldso[p]] = rx[p];
    *(float4*)&ws[0][ldso[p]] = rw[p];
  }
  __syncthreads();

  for (int kb = 0; kb < KDIM; kb += 32) {
    const int  buf  = (kb >> 5) & 1;
    const bool more = (kb + 32) < KDIM;

    // Kick off next tile's global loads; they complete under the WMMAs.
    if (more) {
#pragma unroll
      for (int p = 0; p < 4; ++p) {
        rx[p] = *(const float4*)&x[gxo[p] + kb + 32];
        rw[p] = *(const float4*)&W[gwo[p] + kb + 32];
      }
    }

    const float* xb = &xs[buf][0];
    const float* wb = &ws[buf][0];
#pragma unroll
    for (int ks = 0; ks < 32; ks += 4) {
      const v2f a0 = *(const v2f*)(xb + offA + ks);
      const v2f a1 = *(const v2f*)(xb + offA + 16 * LDSS + ks);
      const v2f b0 = *(const v2f*)(wb + offB + ks);
      const v2f b1 = *(const v2f*)(wb + offB + 16 * LDSS + ks);
      acc[0][0] = __builtin_amdgcn_wmma_f32_16x16x4_f32(
          false, a0, false, b0, (short)0, acc[0][0], false, false);
      acc[0][1] = __builtin_amdgcn_wmma_f32_16x16x4_f32(
          false, a0, false, b1, (short)0, acc[0][1], false, false);
      acc[1][0] = __builtin_amdgcn_wmma_f32_16x16x4_f32(
          false, a1, false, b0, (short)0, acc[1][0], false, false);
      acc[1][1] = __builtin_amdgcn_wmma_f32_16x16x4_f32(
          false, a1, false, b1, (short)0, acc[1][1], false, false);
    }

    // Stage the prefetched tile into the other buffer; one barrier per chunk.
    if (more) {
      float* xn = &xs[buf ^ 1][0];
      float* wn = &ws[buf ^ 1][0];
#pragma unroll
      for (int p = 0; p < 4; ++p) {
        *(float4*)&xn[ldso[p]] = rx[p];
        *(float4*)&wn[ldso[p]] = rw[p];
      }
      __syncthreads();
    }
  }

  // Write back 4 accumulator tiles with bias folded in.
#pragma unroll
  for (int i = 0; i < 2; ++i) {
    const int mrow = m0 + wm + (i << 4) + (half << 3);
#pragma unroll
    for (int j = 0; j < 2; ++j) {
      const int ncol = n0 + wn + (j << 4) + lane16;
      const float bv = bias[ncol];
#pragma unroll
      for (int r = 0; r < 8; ++r) {
        out[(size_t)(mrow + r) * NDIM + ncol] = acc[i][j][r] + bv;
      }
    }
  }
}

// ---------------------------------------------------------------------------
// Launch: inputs per setup_inputs(): x, values, rows, cols, bias.
// d_ws holds the 64MB dense W (re-zeroed every call -> deterministic).
// ---------------------------------------------------------------------------
extern "C" void kernel_launch(void* const* d_in, const int* in_sizes, int n_in,
                              void* d_out, int out_size, void* d_ws, size_t ws_size,
                              hipStream_t stream) {
  const float* x    = (const float*)d_in[0];
  const float* vals = (const float*)d_in[1];
  const int*   rows = (const int*)d_in[2];
  const int*   cols = (const int*)d_in[3];
  const float* bias = (const float*)d_in[4];
  float* outp = (float*)d_out;
  float* W    = (float*)d_ws;

  const int nnz = in_sizes[1];

  // 1) zero W: 4096*4096 floats = 4M float4
  const int n4 = (NDIM * KDIM) / 4;
  sl_zero<<<(n4 + 255) / 256, 256, 0, stream>>>((float4*)W, n4);

  // 2) scatter COO
  sl_scatter<<<(nnz + 255) / 256, 256, 0, stream>>>(vals, rows, cols, W, nnz);

  // 3) GEMM: 64x64 tiles over [4096 x 4096]
  dim3 grid(NDIM / 64, 4096 / 64);
  sl_gemm<<<grid, 128, 0, stream>>>(x, W, bias, outp);
}